// SparseConv3dTestTorch_26963804684442
// MI455X (gfx1250) — compile-verified
//
#include <hip/hip_runtime.h>

typedef __attribute__((ext_vector_type(2))) float v2f;
typedef __attribute__((ext_vector_type(8))) float v8f;

#define CIN  32
#define COUT 32
#define KVOL 27
#define TILE 16
#define WPB  8   // waves (tiles) per 256-thread block

__global__ void zero_f32(float* __restrict__ p, int n) {
  int i = blockIdx.x * 256 + threadIdx.x;
  if (i < n) p[i] = 0.0f;
}

// One wave = one 16-row tile of pairs for one kernel offset k (k = blockIdx.y).
// D(16x32 f32) = A(16x32 gathered feats) x B(32x32 weight[k]) via
// 8 K-steps x 2 N-halves of v_wmma_f32_16x16x4_f32, then scatter-atomic-add.
__global__ __launch_bounds__(256) void spconv_wmma(
    const float* __restrict__ feat,     // [N, 32]
    const float* __restrict__ weight,   // [27, 32, 32] (k, cin, cout)
    const int*   __restrict__ gidx,     // [27, N]
    const int*   __restrict__ sidx,     // [27, N]
    float*       __restrict__ out,      // [num_out, 32]
    int N, int num_out)
{
  __shared__ float wl[CIN * COUT];      // packed (cin/2, cout, 2)

  const int k    = blockIdx.y;
  const int lane = threadIdx.x & 31;
  const int wave = threadIdx.x >> 5;

  // Cooperative LDS fill of weight[k], packed so each B-operand half is one b64:
  // wl[(cin>>1)*64 + cout*2 + (cin&1)] = w[cin][cout]
  {
    const float* wk = weight + (long)k * (CIN * COUT);
    #pragma unroll
    for (int t = 0; t < (CIN * COUT) / 256; ++t) {
      int i = threadIdx.x + t * 256;
      int ci = i >> 5, co = i & 31;
      wl[(ci >> 1) * 64 + co * 2 + (ci & 1)] = wk[i];
    }
  }
  __syncthreads();

  const int base = (blockIdx.x * WPB + wave) * TILE;
  if (base >= N) return;                 // whole-wave exit: EXEC stays all-1s for WMMA

  const int  hi = lane >> 4;             // 0: lanes 0-15 (K 0,1 / M g), 1: lanes 16-31 (K 2,3 / M g+8)
  const int  m  = lane & 15;
  const long kN = (long)k * N;

  // Gather index for row M=m (streaming, read-once -> non-temporal).
  const int   grow = __builtin_nontemporal_load(&gidx[kN + base + m]);
  const float* arow = feat + (long)grow * CIN + hi * 2;
  const v2f*  wl2  = (const v2f*)wl;

  v8f c0 = {};   // columns 0..15
  v8f c1 = {};   // columns 16..31
  #pragma unroll
  for (int s = 0; s < 8; ++s) {
    // A 16x4 slab: lane holds feat[grow][4s + 2*hi .. +1] (features stay hot in L2)
    v2f a  = *(const v2f*)(arow + 4 * s);
    // B 4x16 slabs from LDS: K rows {4s+2hi, 4s+2hi+1}, cols m / 16+m
    v2f b0 = wl2[(2 * s + hi) * 32 + m];
    v2f b1 = wl2[(2 * s + hi) * 32 + 16 + m];
    c0 = __builtin_amdgcn_wmma_f32_16x16x4_f32(false, a, false, b0, (short)0, c0, false, false);
    c1 = __builtin_amdgcn_wmma_f32_16x16x4_f32(false, a, false, b1, (short)0, c1, false, false);
  }

  // Scatter-add: accumulator VGPR g holds row M=g (lanes 0-15) / M=g+8 (lanes 16-31).
  #pragma unroll
  for (int g = 0; g < 8; ++g) {
    const int srow = __builtin_nontemporal_load(&sidx[kN + base + g + hi * 8]);
    if (srow < num_out) {                // drop dummy row (== num_out)
      float* orow = out + (long)srow * COUT + m;
      atomicAdd(orow,      c0[g]);
      atomicAdd(orow + 16, c1[g]);
    }
  }
}

extern "C" void kernel_launch(void* const* d_in, const int* in_sizes, int n_in,
                              void* d_out, int out_size, void* d_ws, size_t ws_size,
                              hipStream_t stream) {
  const float* feat   = (const float*)d_in[0];
  const float* weight = (const float*)d_in[1];
  const int*   gidx   = (const int*)d_in[2];
  const int*   sidx   = (const int*)d_in[3];
  float*       out    = (float*)d_out;

  const int N       = in_sizes[0] / CIN;   // 150000
  const int num_out = out_size / COUT;     // M (d_out excludes the dummy row)

  zero_f32<<<(out_size + 255) / 256, 256, 0, stream>>>(out, out_size);

  const int tiles_per_k  = (N + TILE - 1) / TILE;
  const int blocks_per_k = (tiles_per_k + WPB - 1) / WPB;
  dim3 grid(blocks_per_k, KVOL, 1);
  spconv_wmma<<<grid, 256, 0, stream>>>(feat, weight, gidx, sidx, out, N, num_out);
}